// Block2DGRU_12532714570194
// MI455X (gfx1250) — compile-verified
//
#include <hip/hip_runtime.h>
#include <math.h>

typedef __attribute__((ext_vector_type(2))) float v2f;
typedef __attribute__((ext_vector_type(8))) float v8f;

#define NBK  56
#define LSEQ (NBK * NBK)        // 3136
#define DIM  384
#define NTOK (16 * LSEQ)        // 50176

// ---------------------------------------------------------------------------
// LayerNorm over last dim (384). One block (128 threads) per token row.
// ---------------------------------------------------------------------------
__global__ __launch_bounds__(128) void ln_kernel(const float* __restrict__ X,
                                                 const float* __restrict__ w,
                                                 const float* __restrict__ b,
                                                 float* __restrict__ Y) {
  __shared__ float red[4];
  const size_t row = blockIdx.x;
  const float* x = X + row * DIM;
  float* y = Y + row * DIM;
  const int t = threadIdx.x;

  float v0 = x[t], v1 = x[t + 128], v2 = x[t + 256];
  float s = v0 + v1 + v2;
  #pragma unroll
  for (int o = 16; o; o >>= 1) s += __shfl_xor(s, o, 32);
  if ((t & 31) == 0) red[t >> 5] = s;
  __syncthreads();
  const float mu = (red[0] + red[1] + red[2] + red[3]) * (1.0f / DIM);
  __syncthreads();

  const float d0 = v0 - mu, d1 = v1 - mu, d2 = v2 - mu;
  float q = d0 * d0 + d1 * d1 + d2 * d2;
  #pragma unroll
  for (int o = 16; o; o >>= 1) q += __shfl_xor(q, o, 32);
  if ((t & 31) == 0) red[t >> 5] = q;
  __syncthreads();
  const float var = (red[0] + red[1] + red[2] + red[3]) * (1.0f / DIM);
  const float rstd = rsqrtf(var + 1e-5f);

  y[t]       = d0 * rstd * w[t]       + b[t];
  y[t + 128] = d1 * rstd * w[t + 128] + b[t + 128];
  y[t + 256] = d2 * rstd * w[t + 256] + b[t + 256];
}

// ---------------------------------------------------------------------------
// Depthwise 3x3 conv, NHWC layout [n,56,56,384], zero ("SAME") padding.
// One block per pixel, 384 threads over channels (fully coalesced).
// ---------------------------------------------------------------------------
__global__ __launch_bounds__(384) void dwconv_kernel(const float* __restrict__ X,
                                                     const float* __restrict__ w9,
                                                     const float* __restrict__ bias,
                                                     float* __restrict__ Y) {
  const int c = threadIdx.x;
  const int p = blockIdx.x;            // n*3136 + y*56 + x
  const int nb = p / LSEQ;
  const int yx = p - nb * LSEQ;
  const int y = yx / NBK;
  const int x = yx - y * NBK;

  float wv[9];
  #pragma unroll
  for (int q = 0; q < 9; ++q) wv[q] = w9[c * 9 + q];

  float acc = bias[c];
  #pragma unroll
  for (int dy = 0; dy < 3; ++dy) {
    const int yy = y + dy - 1;
    if (yy < 0 || yy >= NBK) continue;
    #pragma unroll
    for (int dx = 0; dx < 3; ++dx) {
      const int xx = x + dx - 1;
      if (xx < 0 || xx >= NBK) continue;
      acc += X[((size_t)nb * LSEQ + yy * NBK + xx) * DIM + c] * wv[dy * 3 + dx];
    }
  }
  Y[(size_t)p * DIM + c] = acc;
}

// ---------------------------------------------------------------------------
// Pack [w1_hg | w2_hg] -> [384, 3072]
// ---------------------------------------------------------------------------
__global__ void pack_whg_kernel(const float* __restrict__ w1,
                                const float* __restrict__ w2,
                                float* __restrict__ wc) {
  const int i = blockIdx.x * blockDim.x + threadIdx.x;   // over 384*3072
  if (i >= DIM * 3072) return;
  const int k = i / 3072;
  const int j = i - k * 3072;
  wc[i] = (j < 1536) ? w1[k * 1536 + j] : w2[k * 1536 + (j - 1536)];
}

// ---------------------------------------------------------------------------
// minGRU linear recurrence scan. hg: [n, L, 3072] with per-branch
// [hidden(768) | gate(768)] pairs; out: [n, L, 1536]. One warp handles 32
// channels; branch 0 scans forward, branch 1 scans in reverse (== the
// 2D-flip branch of the reference). Software-prefetch next step's loads.
// ---------------------------------------------------------------------------
__global__ __launch_bounds__(256) void scan_kernel(const float* __restrict__ hg,
                                                   float* __restrict__ out) {
  const int gw   = (blockIdx.x * 256 + threadIdx.x) >> 5;   // 0..767
  const int lane = threadIdx.x & 31;
  const int n    = gw / 48;
  const int rem  = gw - n * 48;
  const int br   = rem / 24;
  const int c    = (rem - br * 24) * 32 + lane;

  const float* base  = hg  + (size_t)n * LSEQ * 3072 + (size_t)br * 1536 + c;
  float*       obase = out + (size_t)n * LSEQ * 1536 + (size_t)br * 768  + c;

  const int t0 = br ? (LSEQ - 1) : 0;
  const long long dip = br ? -3072 : 3072;
  const long long dop = br ? -1536 : 1536;
  long long ip = (long long)t0 * 3072;
  long long op = (long long)t0 * 1536;

  float state = 0.0f;
  float h = base[ip];
  float g = base[ip + 768];
  for (int i = 0; i < LSEQ; ++i) {
    float hn = 0.0f, gn = 0.0f;
    if (i + 1 < LSEQ) { hn = base[ip + dip]; gn = base[ip + dip + 768]; }
    const float z     = 1.0f / (1.0f + __expf(-g));                       // sigmoid(gate)
    const float tilde = (h >= 0.0f) ? (h + 0.5f) : (1.0f / (1.0f + __expf(-h)));
    state = (1.0f - z) * state + z * tilde;
    obase[op] = state;
    ip += dip; op += dop; h = hn; g = gn;
  }
}

// ---------------------------------------------------------------------------
// fp32 GEMM via V_WMMA_F32_16X16X4_F32.  D[M,N] = A[M,K] @ W[K,N] (+epilogue)
// Block tile 128x64, BK=16, double-buffered LDS, 8 waves each owning 32x32.
// EPI: 0 = none, 1 = +R, 2 = bias+GELU(exact), 3 = bias + R
// ---------------------------------------------------------------------------
template <int EPI>
__global__ __launch_bounds__(256) void gemm_wmma_f32(const float* __restrict__ A,
                                                     const float* __restrict__ W,
                                                     float* __restrict__ D,
                                                     const float* __restrict__ bias,
                                                     const float* __restrict__ R,
                                                     int M, int K, int N) {
  constexpr int BM = 128, BN = 64, BK = 16;
  __shared__ float sA[2][BK][BM + 1];   // K-major; stride 129 (odd) -> conflict-free
  __shared__ float sB[2][BK][BN + 4];

  const int tid = threadIdx.x;
  const int m0 = blockIdx.y * BM;
  const int n0 = blockIdx.x * BN;

  // Global staging assignments: A -> 8 floats/thread, B -> 4 floats/thread.
  const int ar = tid >> 1, ac = (tid & 1) * 8;
  const int brr = tid >> 4, bcc = (tid & 15) * 4;
  const float* aptr = A + (size_t)(m0 + ar) * K + ac;
  const float* bptr = W + (size_t)brr * N + n0 + bcc;

  float4 ra0, ra1, rb;
  auto fetch = [&](int k0) {
    const float* ap = aptr + k0;
    ra0 = *(const float4*)(ap);
    ra1 = *(const float4*)(ap + 4);
    rb  = *(const float4*)(bptr + (size_t)k0 * N);
  };
  auto stage = [&](int buf) {
    sA[buf][ac + 0][ar] = ra0.x; sA[buf][ac + 1][ar] = ra0.y;
    sA[buf][ac + 2][ar] = ra0.z; sA[buf][ac + 3][ar] = ra0.w;
    sA[buf][ac + 4][ar] = ra1.x; sA[buf][ac + 5][ar] = ra1.y;
    sA[buf][ac + 6][ar] = ra1.z; sA[buf][ac + 7][ar] = ra1.w;
    *(float4*)&sB[buf][brr][bcc] = rb;
  };

  const int lane = tid & 31;
  const int wave = tid >> 5;
  const int wm = (wave & 3) * 32;      // 4 waves down M
  const int wn = (wave >> 2) * 32;     // 2 waves across N
  const int ml = lane & 15;
  const int kh = lane >> 4;            // lane half selects K pairs (ISA A/B layout)

  const v8f zero8 = {0.f, 0.f, 0.f, 0.f, 0.f, 0.f, 0.f, 0.f};
  v8f acc[2][2];
  #pragma unroll
  for (int i = 0; i < 2; ++i)
    #pragma unroll
    for (int j = 0; j < 2; ++j) acc[i][j] = zero8;

  const int nK = K / BK;
  fetch(0); stage(0);
  __syncthreads();

  for (int kt = 0; kt < nK; ++kt) {
    const int cur = kt & 1;
    if (kt + 1 < nK) fetch((kt + 1) * BK);

    #pragma unroll
    for (int kb = 0; kb < BK; kb += 4) {
      v2f a[2], b[2];
      #pragma unroll
      for (int i = 0; i < 2; ++i) {
        // A 16x4 fragment: VGPR0 = K0/K2 (lane halves), VGPR1 = K1/K3
        a[i].x = sA[cur][kb + kh * 2 + 0][wm + i * 16 + ml];
        a[i].y = sA[cur][kb + kh * 2 + 1][wm + i * 16 + ml];
      }
      #pragma unroll
      for (int j = 0; j < 2; ++j) {
        // B 4x16 fragment: VGPR0 = K0/K1 (lane halves), VGPR1 = K2/K3
        b[j].x = sB[cur][kb + kh][wn + j * 16 + ml];
        b[j].y = sB[cur][kb + 2 + kh][wn + j * 16 + ml];
      }
      #pragma unroll
      for (int i = 0; i < 2; ++i)
        #pragma unroll
        for (int j = 0; j < 2; ++j)
          acc[i][j] = __builtin_amdgcn_wmma_f32_16x16x4_f32(
              false, a[i], false, b[j], (short)0, acc[i][j], false, false);
    }

    __syncthreads();
    if (kt + 1 < nK) { stage(cur ^ 1); __syncthreads(); }
  }

  // Epilogue + store. C/D layout: VGPR r -> rows r (lanes 0-15) and r+8 (16-31).
  #pragma unroll
  for (int i = 0; i < 2; ++i) {
    #pragma unroll
    for (int j = 0; j < 2; ++j) {
      #pragma unroll
      for (int r = 0; r < 8; ++r) {
        const int gm = m0 + wm + i * 16 + kh * 8 + r;
        const int gn = n0 + wn + j * 16 + ml;
        const size_t idx = (size_t)gm * N + gn;
        float v = acc[i][j][r];
        if (EPI == 1) {
          v += R[idx];
        } else if (EPI == 2) {
          v += bias[gn];
          v = 0.5f * v * (1.0f + erff(v * 0.70710678118654752f));   // exact GELU
        } else if (EPI == 3) {
          v += bias[gn] + R[idx];
        }
        D[idx] = v;
      }
    }
  }
}

// ---------------------------------------------------------------------------
// Orchestration
// ---------------------------------------------------------------------------
extern "C" void kernel_launch(void* const* d_in, const int* in_sizes, int n_in,
                              void* d_out, int out_size, void* d_ws, size_t ws_size,
                              hipStream_t stream) {
  (void)in_sizes; (void)n_in; (void)out_size; (void)ws_size;

  const float* x       = (const float*)d_in[0];
  const float* norm_w  = (const float*)d_in[1];
  const float* norm_b  = (const float*)d_in[2];
  const float* dw_w    = (const float*)d_in[3];
  const float* dw_b    = (const float*)d_in[4];
  const float* g1_whg  = (const float*)d_in[5];
  const float* g1_wout = (const float*)d_in[6];
  const float* g2_whg  = (const float*)d_in[7];
  const float* g2_wout = (const float*)d_in[8];
  const float* n2_w    = (const float*)d_in[9];
  const float* n2_b    = (const float*)d_in[10];
  const float* p1_w    = (const float*)d_in[11];
  const float* p1_b    = (const float*)d_in[12];
  const float* p2_w    = (const float*)d_in[13];
  const float* p2_b    = (const float*)d_in[14];

  // Workspace layout (bytes, all 16B aligned); buffers are reused over time.
  char* ws = (char*)d_ws;
  float* h_ln   = (float*)(ws + 0);            // 50176*384  = 77,070,336 B (later m_ln)
  float* h_conv = (float*)(ws + 77070336);     // 50176*384              (later h_mid)
  float* hg_cat = (float*)(ws + 154140672);    // 50176*3072 = 616,562,688 B (later m1)
  float* scan_c = (float*)(ws + 770703360);    // 50176*1536 = 308,281,344 B
  float* wc_hg  = (float*)(ws + 1078984704);   // 384*3072   =   4,718,592 B
  float* wc_out = (float*)(ws + 1083703296);   // 1536*384   =   2,359,296 B
  float* h_mid  = h_conv;
  float* m_ln   = h_ln;
  float* m1     = hg_cat;

  // 1. LayerNorm1: x -> h_ln
  ln_kernel<<<NTOK, 128, 0, stream>>>(x, norm_w, norm_b, h_ln);

  // 2. Depthwise 3x3 conv: h_ln -> h_conv
  dwconv_kernel<<<NTOK, 384, 0, stream>>>(h_ln, dw_w, dw_b, h_conv);

  // 3. Pack weights: [w1_hg | w2_hg] and [w1_out ; w2_out]
  pack_whg_kernel<<<(DIM * 3072 + 255) / 256, 256, 0, stream>>>(g1_whg, g2_whg, wc_hg);
  hipMemcpyAsync(wc_out,             g1_wout, (size_t)768 * 384 * 4,
                 hipMemcpyDeviceToDevice, stream);
  hipMemcpyAsync(wc_out + 768 * 384, g2_wout, (size_t)768 * 384 * 4,
                 hipMemcpyDeviceToDevice, stream);

  // 4. hg_cat = h_conv @ wc_hg   [50176,384] x [384,3072]
  gemm_wmma_f32<0><<<dim3(3072 / 64, NTOK / 128), 256, 0, stream>>>(
      h_conv, wc_hg, hg_cat, nullptr, nullptr, NTOK, 384, 3072);

  // 5. Bidirectional minGRU scans (branch 1 fwd, branch 2 rev == flipped branch)
  scan_kernel<<<96, 256, 0, stream>>>(hg_cat, scan_c);

  // 6. h_mid = scan_c @ wc_out + x   (x1 + x2 + residual)
  gemm_wmma_f32<1><<<dim3(384 / 64, NTOK / 128), 256, 0, stream>>>(
      scan_c, wc_out, h_mid, nullptr, x, NTOK, 1536, 384);

  // 7. LayerNorm2: h_mid -> m_ln
  ln_kernel<<<NTOK, 128, 0, stream>>>(h_mid, n2_w, n2_b, m_ln);

  // 8. m1 = GELU(m_ln @ p1_w + p1_b)
  gemm_wmma_f32<2><<<dim3(1536 / 64, NTOK / 128), 256, 0, stream>>>(
      m_ln, p1_w, m1, p1_b, nullptr, NTOK, 384, 1536);

  // 9. out = m1 @ p2_w + p2_b + h_mid
  gemm_wmma_f32<3><<<dim3(384 / 64, NTOK / 128), 256, 0, stream>>>(
      m1, p2_w, (float*)d_out, p2_b, h_mid, NTOK, 1536, 384);
}